// GCNNet_71347996721293
// MI455X (gfx1250) — compile-verified
//
#include <hip/hip_runtime.h>

#define DIM 128
#define TILES_N 8  // 128/16 output column tiles per wave

typedef float v2f __attribute__((ext_vector_type(2)));
typedef float v8f __attribute__((ext_vector_type(8)));

// ---------------- CSR build: count / dinv / scan / fill ----------------

__global__ __launch_bounds__(256) void k_zero(int* __restrict__ a, int n) {
    int i = blockIdx.x * 256 + threadIdx.x;
    if (i < n) a[i] = 0;
}

__global__ __launch_bounds__(256) void k_count(const int* __restrict__ dst,
                                               int* __restrict__ cnt, int e) {
    int i = blockIdx.x * 256 + threadIdx.x;
    if (i < e) atomicAdd(&cnt[dst[i]], 1);
}

__global__ __launch_bounds__(256) void k_dinv(const int* __restrict__ cnt,
                                              float* __restrict__ dinv, int n) {
    int i = blockIdx.x * 256 + threadIdx.x;
    if (i < n) dinv[i] = rsqrtf((float)cnt[i] + 1.0f);  // +1 self-loop; always > 0
}

__global__ __launch_bounds__(256) void k_partial(const int* __restrict__ cnt,
                                                 int* __restrict__ partial, int n) {
    __shared__ int sh[256];
    int tid = threadIdx.x;
    int i = blockIdx.x * 256 + tid;
    sh[tid] = (i < n) ? cnt[i] : 0;
    __syncthreads();
    for (int s = 128; s > 0; s >>= 1) {
        if (tid < s) sh[tid] += sh[tid + s];
        __syncthreads();
    }
    if (tid == 0) partial[blockIdx.x] = sh[0];
}

// single block; requires nb <= 512 (N <= 131072)
__global__ __launch_bounds__(512) void k_scan_partials(const int* __restrict__ partial,
                                                       int* __restrict__ block_off,
                                                       int* __restrict__ row_ptr,
                                                       int nb, int n) {
    __shared__ int sh[512];
    int tid = threadIdx.x;
    int v = (tid < nb) ? partial[tid] : 0;
    sh[tid] = v;
    __syncthreads();
    for (int off = 1; off < 512; off <<= 1) {
        int t = (tid >= off) ? sh[tid - off] : 0;
        __syncthreads();
        sh[tid] += t;
        __syncthreads();
    }
    if (tid < nb) block_off[tid] = sh[tid] - v;   // exclusive
    if (tid == 511) row_ptr[n] = sh[511];         // total = E
}

__global__ __launch_bounds__(256) void k_scan_chunks(const int* __restrict__ cnt,
                                                     const int* __restrict__ block_off,
                                                     int* __restrict__ row_ptr,
                                                     int* __restrict__ cursor, int n) {
    __shared__ int sh[256];
    int tid = threadIdx.x;
    int i = blockIdx.x * 256 + tid;
    int v = (i < n) ? cnt[i] : 0;
    sh[tid] = v;
    __syncthreads();
    for (int off = 1; off < 256; off <<= 1) {
        int t = (tid >= off) ? sh[tid - off] : 0;
        __syncthreads();
        sh[tid] += t;
        __syncthreads();
    }
    if (i < n) {
        int excl = block_off[blockIdx.x] + sh[tid] - v;
        row_ptr[i] = excl;
        cursor[i]  = excl;
    }
}

__global__ __launch_bounds__(256) void k_fill(const int* __restrict__ src,
                                              const int* __restrict__ dst,
                                              const float* __restrict__ dinv,
                                              int* __restrict__ cursor,
                                              int* __restrict__ src_sorted,
                                              float* __restrict__ w_sorted, int e) {
    int i = blockIdx.x * 256 + threadIdx.x;
    if (i < e) {
        int s = src[i], d = dst[i];
        int pos = atomicAdd(&cursor[d], 1);
        src_sorted[pos] = s;
        w_sorted[pos]   = dinv[s] * dinv[d];  // weight computed once, reused both layers
    }
}

// ---------------- GEMM: H = X @ W  (WMMA f32 16x16x4) ----------------

__global__ __launch_bounds__(256) void k_gemm(
    const float* __restrict__ X, const float* __restrict__ W,
    float* __restrict__ H, int N)
{
    // W fragments pre-swizzled: Wf[kt][nt][lane] = {W[k][n], W[k+1][n]}
    // with k = kt*4 + 2*(lane>>4), n = nt*16 + (lane&15)
    __shared__ float2 Wf[32][TILES_N][32];

    const int tid = threadIdx.x;
    for (int i = tid; i < 32 * TILES_N * 32; i += 256) {
        int kt   = i >> 8;
        int rem  = i & 255;
        int nt   = rem >> 5;
        int lane = rem & 31;
        int k = kt * 4 + ((lane >> 4) << 1);
        int n = nt * 16 + (lane & 15);
        Wf[kt][nt][lane] = make_float2(W[k * DIM + n], W[(k + 1) * DIM + n]);
    }
    __syncthreads();

    const int lane = tid & 31;
    const int wave = tid >> 5;
    const int row0 = blockIdx.x * 128 + wave * 16;
    if (row0 >= N) return;  // wave-uniform; active waves keep EXEC all-ones

    v8f c[TILES_N] = {};

    const int arow = row0 + (lane & 15);
    const float* aptr = X + (size_t)arow * DIM + ((lane >> 4) << 1);

    #pragma unroll 4
    for (int kt = 0; kt < 32; ++kt) {
        float2 av = *(const float2*)(aptr + kt * 4);
        v2f A; A.x = av.x; A.y = av.y;
        #pragma unroll
        for (int nt = 0; nt < TILES_N; ++nt) {
            float2 bv = Wf[kt][nt][lane];
            v2f B; B.x = bv.x; B.y = bv.y;
            c[nt] = __builtin_amdgcn_wmma_f32_16x16x4_f32(
                false, A, false, B, (short)0, c[nt], false, false);
        }
    }

    // C/D layout: row = row0 + v + 8*(lane>>4), col = nt*16 + (lane&15)
    const int col = lane & 15;
    const int rb  = row0 + ((lane >> 4) << 3);
    #pragma unroll
    for (int v = 0; v < 8; ++v) {
        size_t base = (size_t)(rb + v) * DIM;
        #pragma unroll
        for (int nt = 0; nt < TILES_N; ++nt) {
            H[base + nt * 16 + col] = c[nt][v];
        }
    }
}

// ---------------- gather aggregation (one wave per node, no atomics) --------
// out = [relu]( X + bias + dinv^2 * H[node] + sum_j w_j * H[src_j] )
template <bool RELU>
__global__ __launch_bounds__(256) void k_gather_agg(
    const float* __restrict__ X, const float* __restrict__ H,
    const float* __restrict__ bias, const float* __restrict__ dinv,
    const int* __restrict__ row_ptr, const int* __restrict__ src_sorted,
    const float* __restrict__ w_sorted, float* __restrict__ out, int N)
{
    int node = (blockIdx.x * 256 + threadIdx.x) >> 5;
    int lane = threadIdx.x & 31;
    if (node >= N) return;

    int beg = row_ptr[node];
    int end = row_ptr[node + 1];
    size_t co = (size_t)lane * 4;

    float4 acc = make_float4(0.f, 0.f, 0.f, 0.f);
    int j = beg;
    for (; j + 1 < end; j += 2) {
        int   s0 = src_sorted[j],   s1 = src_sorted[j + 1];
        float w0 = w_sorted[j],     w1 = w_sorted[j + 1];
        float4 v0 = *(const float4*)(H + (size_t)s0 * DIM + co);
        float4 v1 = *(const float4*)(H + (size_t)s1 * DIM + co);
        acc.x += w0 * v0.x + w1 * v1.x;
        acc.y += w0 * v0.y + w1 * v1.y;
        acc.z += w0 * v0.z + w1 * v1.z;
        acc.w += w0 * v0.w + w1 * v1.w;
    }
    if (j < end) {
        int   s0 = src_sorted[j];
        float w0 = w_sorted[j];
        float4 v0 = *(const float4*)(H + (size_t)s0 * DIM + co);
        acc.x += w0 * v0.x;
        acc.y += w0 * v0.y;
        acc.z += w0 * v0.z;
        acc.w += w0 * v0.w;
    }

    float di  = dinv[node];
    float di2 = di * di;
    size_t base = (size_t)node * DIM + co;
    float4 hv = *(const float4*)(H + base);
    float4 xv = *(const float4*)(X + base);
    float4 bv = *(const float4*)(bias + co);

    float4 r;
    r.x = xv.x + bv.x + di2 * hv.x + acc.x;
    r.y = xv.y + bv.y + di2 * hv.y + acc.y;
    r.z = xv.z + bv.z + di2 * hv.z + acc.z;
    r.w = xv.w + bv.w + di2 * hv.w + acc.w;
    if (RELU) {
        r.x = fmaxf(r.x, 0.f);
        r.y = fmaxf(r.y, 0.f);
        r.z = fmaxf(r.z, 0.f);
        r.w = fmaxf(r.w, 0.f);
    }
    *(float4*)(out + base) = r;
}

// ---------------- launcher ------------------------------------------------------

extern "C" void kernel_launch(void* const* d_in, const int* in_sizes, int n_in,
                              void* d_out, int out_size, void* d_ws, size_t ws_size,
                              hipStream_t stream) {
    const float* x  = (const float*)d_in[0];
    const int*   ei = (const int*)d_in[1];   // edge_index [2, E]
    const float* W1 = (const float*)d_in[2];
    const float* b1 = (const float*)d_in[3];
    const float* W2 = (const float*)d_in[4];
    const float* b2 = (const float*)d_in[5];

    const int N = in_sizes[0] / DIM;
    const int E = in_sizes[1] / 2;
    const int* src = ei;
    const int* dst = ei + E;

    const size_t nf = (size_t)N * DIM;
    const int NB = (N + 255) / 256;  // scan chunks; must be <= 512

    // workspace carve-out (256B-aligned regions)
    char* wp = (char*)d_ws;
    auto carve = [&](size_t bytes) -> void* {
        void* r = (void*)wp;
        wp += (bytes + 255) & ~(size_t)255;
        return r;
    };
    float* dinv       = (float*)carve((size_t)N * 4);
    int*   cnt        = (int*)  carve((size_t)N * 4);
    int*   row_ptr    = (int*)  carve((size_t)(N + 1) * 4);
    int*   cursor     = (int*)  carve((size_t)N * 4);
    int*   partial    = (int*)  carve((size_t)NB * 4);
    int*   block_off  = (int*)  carve((size_t)NB * 4);
    int*   src_sorted = (int*)  carve((size_t)E * 4);
    float* w_sorted   = (float*)carve((size_t)E * 4);
    float* bufH       = (float*)carve(nf * 4);
    float* bufR       = (float*)carve(nf * 4);
    float* out        = (float*)d_out;

    dim3 blk(256);
    int gn  = (N + 255) / 256;
    int ge  = (E + 255) / 256;
    int gg  = (N + 127) / 128;
    int gw  = (int)(((long long)N * 32 + 255) / 256);  // one wave per node

    // ---- CSR-by-destination build (once; shared by both layers) ----
    k_zero         <<<gn, blk, 0, stream>>>(cnt, N);
    k_count        <<<ge, blk, 0, stream>>>(dst, cnt, E);
    k_dinv         <<<gn, blk, 0, stream>>>(cnt, dinv, N);
    k_partial      <<<NB, blk, 0, stream>>>(cnt, partial, N);
    k_scan_partials<<<1, dim3(512), 0, stream>>>(partial, block_off, row_ptr, NB, N);
    k_scan_chunks  <<<NB, blk, 0, stream>>>(cnt, block_off, row_ptr, cursor, N);
    k_fill         <<<ge, blk, 0, stream>>>(src, dst, dinv, cursor, src_sorted, w_sorted, E);

    // ---- layer 1: h1 = x@W1 ; hrelu = relu(x + b1 + agg(h1)) ----
    k_gemm              <<<gg, blk, 0, stream>>>(x, W1, bufH, N);
    k_gather_agg<true>  <<<gw, blk, 0, stream>>>(x, bufH, b1, dinv, row_ptr,
                                                 src_sorted, w_sorted, bufR, N);

    // ---- layer 2: h2 = hrelu@W2 ; out = hrelu + b2 + agg(h2) ----
    k_gemm              <<<gg, blk, 0, stream>>>(bufR, W2, bufH, N);
    k_gather_agg<false> <<<gw, blk, 0, stream>>>(bufR, bufH, b2, dinv, row_ptr,
                                                 src_sorted, w_sorted, out, N);
}